// AR_21019569947274
// MI455X (gfx1250) — compile-verified
//
#include <hip/hip_runtime.h>
#include <hip/hip_bf16.h>

typedef __attribute__((ext_vector_type(2))) float v2f;
typedef __attribute__((ext_vector_type(8))) float v8f;
typedef long long ll;

#define WAVES 4   // 4 waves (128 threads) per block; one 32x32 tile per wave

#define WMMA_F32(A_, B_, C_) \
  __builtin_amdgcn_wmma_f32_16x16x4_f32(false, (A_), false, (B_), (short)0, (C_), false, false)

// ---------------------------------------------------------------------------
// Generic batched GEMM:  C[z] = alpha * A[z] @ W[z] (+ bias) (+ resid)
// f32 WMMA 16x16x4, 2x2 register blocking (32x32 output per wave, 4 indep
// accumulator chains).  Branch-free unswitched inner loops.
// A-frag (ISA 7.12.2, 32-bit A 16x4): lane {m=l&15, kh=l>>4}; float2 = K(2kh..2kh+1).
// B-frag (4x16): lane holds col n=l&15, rows 2kh / 2kh+1.
// C layout: vgpr r: lanes0-15 -> M=r, lanes16-31 -> M=8+r, N=l&15.
// zdiv splits z=(zhi,zlo) for attention (batch, head) indexing.
// Kvalid/Nvalid clamp W-side loads for the padded-N scores and padded-K P@V paths; the A-side
// pad columns are exact zeros so clamped W rows contribute nothing.
// ---------------------------------------------------------------------------
__global__ __launch_bounds__(32 * WAVES) void gemm_wmma_f32(
    const float* __restrict__ A, int lda, ll sAhi, ll sAlo,
    const float* __restrict__ W, int ldw, ll sWhi, ll sWlo, int wtrans,
    float* __restrict__ C, int ldc, ll sChi, ll sClo,
    const float* __restrict__ bias, const float* __restrict__ resid,
    int M, int N, int K, int Kvalid, int Nvalid, float alpha, int rowmap,
    int zdiv)
{
  int lane = threadIdx.x & 31;
  int wave = threadIdx.x >> 5;
  int Mtiles = (M + 15) >> 4;
  int MPairs = (Mtiles + 1) >> 1;
  int mp = blockIdx.y * WAVES + wave;
  if (mp >= MPairs) return;                 // wave-uniform exit: EXEC stays full
  int z = blockIdx.z;
  int zh = z / zdiv, zl = z % zdiv;

  const float* Az = A + zh * sAhi + zl * sAlo;
  const float* Wz = W + zh * sWhi + zl * sWlo;
  float*       Cz = C + zh * sChi + zl * sClo;
  const float* Rz = resid ? (resid + zh * sChi + zl * sClo) : nullptr;

  int m0 = mp << 5;
  int n0 = blockIdx.x << 5;
  int mr = lane & 15;
  int kh = lane >> 4;                       // 0 or 1

  int mA0 = m0 + mr;       if (mA0 > M - 1) mA0 = M - 1;
  int mA1 = m0 + 16 + mr;  if (mA1 > M - 1) mA1 = M - 1;
  int nB0 = n0 + mr;       if (nB0 > Nvalid - 1) nB0 = Nvalid - 1;
  int nB1 = n0 + 16 + mr;  if (nB1 > Nvalid - 1) nB1 = Nvalid - 1;

  const float* a0p = Az + (ll)mA0 * lda + 2 * kh;
  const float* a1p = Az + (ll)mA1 * lda + 2 * kh;

  v8f acc00 = {0.f,0.f,0.f,0.f,0.f,0.f,0.f,0.f};
  v8f acc01 = acc00, acc10 = acc00, acc11 = acc00;

  int kfast = Kvalid & ~3;                  // full-speed region (no clamping)
  if (wtrans) {
    const float* b0p = Wz + (ll)nB0 * ldw + 2 * kh;
    const float* b1p = Wz + (ll)nB1 * ldw + 2 * kh;
    for (int kk = 0; kk < kfast; kk += 4) {
      v2f a0 = *(const v2f*)a0p;  a0p += 4;
      v2f a1 = *(const v2f*)a1p;  a1p += 4;
      v2f b0 = *(const v2f*)b0p;  b0p += 4;
      v2f b1 = *(const v2f*)b1p;  b1p += 4;
      acc00 = WMMA_F32(a0, b0, acc00);
      acc10 = WMMA_F32(a1, b0, acc10);
      acc01 = WMMA_F32(a0, b1, acc01);
      acc11 = WMMA_F32(a1, b1, acc11);
    }
  } else {
    const float* b0p = Wz + (ll)(2 * kh) * ldw + nB0;
    const float* b1p = Wz + (ll)(2 * kh) * ldw + nB1;
    ll step = (ll)4 * ldw;
    for (int kk = 0; kk < kfast; kk += 4) {
      v2f a0 = *(const v2f*)a0p;  a0p += 4;
      v2f a1 = *(const v2f*)a1p;  a1p += 4;
      v2f b0, b1;
      b0.x = b0p[0]; b0.y = b0p[ldw];
      b1.x = b1p[0]; b1.y = b1p[ldw];
      b0p += step; b1p += step;
      acc00 = WMMA_F32(a0, b0, acc00);
      acc10 = WMMA_F32(a1, b0, acc10);
      acc01 = WMMA_F32(a0, b1, acc01);
      acc11 = WMMA_F32(a1, b1, acc11);
    }
  }
  // Tail: K padded past Kvalid (P@V, K=272 valid 257). A pad columns are zero,
  // so only the W row index is clamped; the zero-A product is exact.
  for (int kk = kfast; kk < K; kk += 4) {
    int k0 = kk + 2 * kh;
    int k1 = k0 + 1;
    int c0 = (k0 > Kvalid - 1) ? (Kvalid - 1) : k0;
    int c1 = (k1 > Kvalid - 1) ? (Kvalid - 1) : k1;
    v2f a0; a0.x = Az[(ll)mA0 * lda + k0]; a0.y = Az[(ll)mA0 * lda + k1];
    v2f a1; a1.x = Az[(ll)mA1 * lda + k0]; a1.y = Az[(ll)mA1 * lda + k1];
    v2f b0, b1;
    if (wtrans) {
      b0.x = Wz[(ll)nB0 * ldw + c0]; b0.y = Wz[(ll)nB0 * ldw + c1];
      b1.x = Wz[(ll)nB1 * ldw + c0]; b1.y = Wz[(ll)nB1 * ldw + c1];
    } else {
      b0.x = Wz[(ll)c0 * ldw + nB0]; b0.y = Wz[(ll)c1 * ldw + nB0];
      b1.x = Wz[(ll)c0 * ldw + nB1]; b1.y = Wz[(ll)c1 * ldw + nB1];
    }
    acc00 = WMMA_F32(a0, b0, acc00);
    acc10 = WMMA_F32(a1, b0, acc10);
    acc01 = WMMA_F32(a0, b1, acc01);
    acc11 = WMMA_F32(a1, b1, acc11);
  }

  // -------- store 4 tiles with edge guards --------
  int nO0 = n0 + (lane & 15);
  int nO1 = nO0 + 16;
  int mB0 = m0 + 8 * kh;
  int mB1 = mB0 + 16;
  float bv0 = (bias && nO0 < N) ? bias[nO0] : 0.f;
  float bv1 = (bias && nO1 < N) ? bias[nO1] : 0.f;
#pragma unroll
  for (int r = 0; r < 8; ++r) {
#pragma unroll
    for (int t = 0; t < 4; ++t) {
      int m = ((t & 1) ? mB1 : mB0) + r;
      int n = (t & 2) ? nO1 : nO0;
      float av = (t == 0) ? acc00[r] : (t == 1) ? acc10[r]
               : (t == 2) ? acc01[r] : acc11[r];
      if (m < M && n < N) {
        ll orow;
        if (rowmap == 1)      orow = (ll)m + (m >> 8) + 1;  // patch: skip CLS
        else if (rowmap == 2) orow = (ll)m * 257;           // cond -> CLS slots
        else                  orow = m;
        float v = alpha * av + ((t & 2) ? bv1 : bv0);
        if (Rz) v += Rz[orow * ldc + n];
        Cz[orow * ldc + n] = v;
      }
    }
  }
}

// ------------------------- row kernels (wave per row) -----------------------
__global__ void rmsnorm_kernel(const float* __restrict__ x,
                               const float* __restrict__ w,
                               float* __restrict__ out, int rows, int D)
{
  int row = blockIdx.x * (blockDim.x >> 5) + (threadIdx.x >> 5);
  int lane = threadIdx.x & 31;
  if (row >= rows) return;
  const float* xr = x + (ll)row * D;
  float ss = 0.f;
  for (int i = lane; i < D; i += 32) { float v = xr[i]; ss += v * v; }
  for (int o = 16; o > 0; o >>= 1) ss += __shfl_xor(ss, o, 32);
  float r = rsqrtf(ss / (float)D + 1e-5f);
  float* orow = out + (ll)row * D;
  for (int i = lane; i < D; i += 32) orow[i] = xr[i] * r * w[i];
}

__global__ void embed_ln_kernel(const float* __restrict__ hp,
                                const float* __restrict__ pos,
                                const float* __restrict__ g,
                                const float* __restrict__ bt,
                                float* __restrict__ out, int rows, int S, int D)
{
  int row = blockIdx.x * (blockDim.x >> 5) + (threadIdx.x >> 5);
  int lane = threadIdx.x & 31;
  if (row >= rows) return;
  int s = row % S;
  const float* x = hp + (ll)row * D;
  const float* p = pos + (ll)s * D;
  float s1 = 0.f, s2 = 0.f;
  for (int i = lane; i < D; i += 32) { float v = x[i] + p[i]; s1 += v; s2 += v * v; }
  for (int o = 16; o > 0; o >>= 1) {
    s1 += __shfl_xor(s1, o, 32);
    s2 += __shfl_xor(s2, o, 32);
  }
  float mean = s1 / (float)D;
  float var = s2 / (float)D - mean * mean;
  float r = rsqrtf(var + 1e-6f);
  float* orow = out + (ll)row * D;
  for (int i = lane; i < D; i += 32) {
    float v = x[i] + p[i];
    orow[i] = (v - mean) * r * g[i] + bt[i];
  }
}

__global__ void freqs_kernel(float* __restrict__ fc)   // fc[257][32][2]
{
  int idx = blockIdx.x * blockDim.x + threadIdx.x;
  if (idx >= 257 * 32) return;
  int s = idx >> 5, j = idx & 31;
  float c = 0.f, sn = 0.f;                 // CLS row = zeros (matches reference)
  if (s > 0) {
    int p = s - 1;
    int i = (j < 16) ? j : (j - 16);
    float t = (j < 16) ? (float)(p >> 4) : (float)(p & 15);
    float fr = powf(10000.0f, -((float)(2 * i)) / 32.0f);
    float f = t * fr;
    c = cosf(f); sn = sinf(f);
  }
  fc[idx * 2] = c; fc[idx * 2 + 1] = sn;
}

__global__ void rope_kernel(float* __restrict__ qkv,
                            const float* __restrict__ fc, int B, int S)
{
  int idx = blockIdx.x * blockDim.x + threadIdx.x;
  int total = B * S * 16 * 32;
  if (idx >= total) return;
  int j = idx & 31;
  int t = idx >> 5;
  int hh = t & 15; t >>= 4;
  int s = t % S, b = t / S;
  float c = fc[(s * 32 + j) * 2], sn = fc[(s * 32 + j) * 2 + 1];
  size_t base = ((size_t)(b * S + s)) * 3072 + (size_t)hh * 64 + 2 * j;
  float x0 = qkv[base], x1 = qkv[base + 1];                       // Q
  qkv[base] = x0 * c - x1 * sn; qkv[base + 1] = x1 * c + x0 * sn;
  base += 1024;                                                   // K
  x0 = qkv[base]; x1 = qkv[base + 1];
  qkv[base] = x0 * c - x1 * sn; qkv[base + 1] = x1 * c + x0 * sn;
}

__global__ void softmax_kernel(float* __restrict__ P, int rows) // ld = 272
{
  int row = blockIdx.x * (blockDim.x >> 5) + (threadIdx.x >> 5);
  int lane = threadIdx.x & 31;
  if (row >= rows) return;
  int q = row % 257;                        // causal: attend cols 0..q
  float* p = P + (size_t)row * 272;
  float mx = -3.4e38f;
  for (int j = lane; j <= q; j += 32) mx = fmaxf(mx, p[j]);
  for (int o = 16; o > 0; o >>= 1) mx = fmaxf(mx, __shfl_xor(mx, o, 32));
  float sum = 0.f;
  for (int j = lane; j <= q; j += 32) { float e = expf(p[j] - mx); p[j] = e; sum += e; }
  for (int o = 16; o > 0; o >>= 1) sum += __shfl_xor(sum, o, 32);
  float inv = 1.0f / sum;
  for (int j = lane; j < 272; j += 32) p[j] = (j <= q) ? p[j] * inv : 0.f;
}

__global__ void silu_mul_kernel(float* __restrict__ u1,
                                const float* __restrict__ u3, size_t n)
{
  size_t idx = (size_t)blockIdx.x * blockDim.x + threadIdx.x;
  if (idx >= n) return;
  float a = u1[idx];
  u1[idx] = (a / (1.0f + expf(-a))) * u3[idx];
}

// ------------------------------- launcher ----------------------------------
static inline void gemm(hipStream_t st,
    const float* A, int lda, ll sAhi, ll sAlo,
    const float* W, int ldw, ll sWhi, ll sWlo, int wtrans,
    float* C, int ldc, ll sChi, ll sClo,
    const float* bias, const float* resid,
    int M, int N, int K, int Kvalid, int Nvalid, float alpha, int rowmap,
    int Z, int zdiv)
{
  int Mtiles = (M + 15) / 16;
  int MPairs = (Mtiles + 1) / 2;
  int NPairs = ((N + 15) / 16 + 1) / 2;
  dim3 grid((unsigned)NPairs,
            (unsigned)((MPairs + WAVES - 1) / WAVES),
            (unsigned)Z);
  gemm_wmma_f32<<<grid, 32 * WAVES, 0, st>>>(
      A, lda, sAhi, sAlo, W, ldw, sWhi, sWlo, wtrans,
      C, ldc, sChi, sClo, bias, resid,
      M, N, K, Kvalid, Nvalid, alpha, rowmap, zdiv);
}

extern "C" void kernel_launch(void* const* d_in, const int* in_sizes, int n_in,
                              void* d_out, int out_size, void* d_ws, size_t ws_size,
                              hipStream_t stream)
{
  (void)in_sizes; (void)n_in; (void)out_size; (void)ws_size;
  const float* x       = (const float*)d_in[0];
  const float* cond    = (const float*)d_in[1];
  const float* patch_w = (const float*)d_in[2];
  const float* patch_b = (const float*)d_in[3];
  const float* ln_g    = (const float*)d_in[4];
  const float* ln_b    = (const float*)d_in[5];
  const float* pos     = (const float*)d_in[6];
  const float* cond_w  = (const float*)d_in[7];
  const float* cond_b  = (const float*)d_in[8];
  const float* wqkv    = (const float*)d_in[9];
  const float* wo      = (const float*)d_in[10];
  const float* w1      = (const float*)d_in[11];
  const float* w2      = (const float*)d_in[12];
  const float* w3      = (const float*)d_in[13];
  const float* anw     = (const float*)d_in[14];
  const float* fnw     = (const float*)d_in[15];

  const int B = 32, S = 257, D = 1024, H = 16, HID = 2816, NB = 12;
  const int SP = 272;                       // S padded to 16-multiple for WMMA K
  const int Mrows = B * S;                  // 8224 = 16 * 514

  float* ws = (float*)d_ws;
  size_t off = 0;
  float* h   = ws + off; off += (size_t)Mrows * D;       // residual stream
  float* a   = ws + off; off += (size_t)Mrows * D;       // norm output / hpre
  float* qkv = ws + off; off += (size_t)Mrows * 3 * D;
  float* o   = ws + off; off += (size_t)Mrows * D;       // attn output
  float* P   = ws + off; off += (size_t)B * H * S * SP;  // softmax probs
  float* u1  = ws + off; off += (size_t)Mrows * HID;
  float* u3  = ws + off; off += (size_t)Mrows * HID;
  float* fc  = ws + off; off += (size_t)S * 64;          // cos/sin cache

  // freqs_cis (tiny)
  freqs_kernel<<<(S * 32 + 255) / 256, 256, 0, stream>>>(fc);

  // patch embed -> rows b*257+1+s ; cond embed -> rows b*257 (into `a` = hpre)
  gemm(stream, x, 768, 0, 0, patch_w, D, 0, 0, 0, a, D, 0, 0,
       patch_b, nullptr, B * 256, D, 768, 768, D, 1.f, 1, 1, 1);
  gemm(stream, cond, D, 0, 0, cond_w, D, 0, 0, 0, a, D, 0, 0,
       cond_b, nullptr, B, D, D, D, D, 1.f, 2, 1, 1);
  // + pos_embed, LayerNorm -> h
  embed_ln_kernel<<<(Mrows + 3) / 4, 128, 0, stream>>>(a, pos, ln_g, ln_b, h,
                                                       Mrows, S, D);

  for (int blk = 0; blk < NB; ++blk) {
    const float* wqkv_b = wqkv + (size_t)blk * D * 3 * D;
    const float* wo_b   = wo   + (size_t)blk * D * D;
    const float* w1_b   = w1   + (size_t)blk * D * HID;
    const float* w2_b   = w2   + (size_t)blk * HID * D;
    const float* w3_b   = w3   + (size_t)blk * D * HID;

    // a = rmsnorm(h) * attn_norm_w
    rmsnorm_kernel<<<(Mrows + 3) / 4, 128, 0, stream>>>(h, anw + blk * D, a,
                                                        Mrows, D);
    // qkv = a @ wqkv
    gemm(stream, a, D, 0, 0, wqkv_b, 3 * D, 0, 0, 0, qkv, 3 * D, 0, 0,
         nullptr, nullptr, Mrows, 3 * D, D, D, 3 * D, 1.f, 0, 1, 1);
    // RoPE on q,k (in place)
    rope_kernel<<<(B * S * H * 32 + 255) / 256, 256, 0, stream>>>(qkv, fc, B, S);
    // scores: P[b,h] = (1/8) * Q @ K^T   (batched z = b*H + h)
    gemm(stream, qkv, 3 * D, (ll)S * 3 * D, 64,
         qkv + D, 3 * D, (ll)S * 3 * D, 64, 1,
         P, SP, (ll)H * S * SP, (ll)S * SP, nullptr, nullptr,
         S, SP, 64, 64, S, 0.125f, 0, B * H, H);
    // causal softmax over rows of P (zero-fills the 257..271 pad)
    softmax_kernel<<<(B * H * S + 3) / 4, 128, 0, stream>>>(P, B * H * S);
    // o[b,h] = P @ V  (K padded to 272; pad cols of P are exact zeros)
    gemm(stream, P, SP, (ll)H * S * SP, (ll)S * SP,
         qkv + 2 * D, 3 * D, (ll)S * 3 * D, 64, 0,
         o, D, (ll)S * D, 64, nullptr, nullptr,
         S, 64, SP, S, 64, 1.f, 0, B * H, H);
    // h += o @ wo (in place residual)
    gemm(stream, o, D, 0, 0, wo_b, D, 0, 0, 0, h, D, 0, 0,
         nullptr, h, Mrows, D, D, D, D, 1.f, 0, 1, 1);
    // a = rmsnorm(h) * ffn_norm_w
    rmsnorm_kernel<<<(Mrows + 3) / 4, 128, 0, stream>>>(h, fnw + blk * D, a,
                                                        Mrows, D);
    // u1 = a @ w1 ; u3 = a @ w3
    gemm(stream, a, D, 0, 0, w1_b, HID, 0, 0, 0, u1, HID, 0, 0,
         nullptr, nullptr, Mrows, HID, D, D, HID, 1.f, 0, 1, 1);
    gemm(stream, a, D, 0, 0, w3_b, HID, 0, 0, 0, u3, HID, 0, 0,
         nullptr, nullptr, Mrows, HID, D, D, HID, 1.f, 0, 1, 1);
    // u1 = silu(u1) * u3
    {
      size_t n = (size_t)Mrows * HID;
      silu_mul_kernel<<<(unsigned)((n + 255) / 256), 256, 0, stream>>>(u1, u3, n);
    }
    // h += u1 @ w2 ; last block writes straight into d_out
    float* cdst = (blk == NB - 1) ? (float*)d_out : h;
    gemm(stream, u1, HID, 0, 0, w2_b, D, 0, 0, 0, cdst, D, 0, 0,
         nullptr, h, Mrows, D, HID, HID, D, 1.f, 0, 1, 1);
  }
}